// SSIMLoss_80358838108887
// MI455X (gfx1250) — compile-verified
//
#include <hip/hip_runtime.h>

typedef float v2f __attribute__((ext_vector_type(2)));
typedef float v8f __attribute__((ext_vector_type(8)));

namespace {
constexpr int kH = 512, kW = 512, kN = 32;
constexpr int kTilesPerImg = (kH / 16) * (kW / 16);   // 1024
constexpr int kTiles = kTilesPerImg * kN;             // 32768
constexpr int kWavesPerBlock = 4;
constexpr int kBlockThreads = kWavesPerBlock * 32;    // 128 = 4 wave32
constexpr int kBlocks = kTiles / kWavesPerBlock;      // 8192
constexpr float kC1 = 1.0e-4f;                        // 0.01^2
constexpr float kC2 = 9.0e-4f;                        // 0.03^2
constexpr float kInvTotal = 1.0f / (float(kN) * kH * kW);
}

// 11-tap Gaussian (sigma = 1.5), normalized in f32. d in [0,10] -> g[d], else 0.
// Branchless selects only (no dynamically indexed array -> no scratch spills).
__device__ __forceinline__ float gtap(int d) {
    int e = d - 5; e = e < 0 ? -e : e;
    float w = (e == 0) ? 0.26601171f
            : (e == 1) ? 0.21300553f
            : (e == 2) ? 0.10936069f
            : (e == 3) ? 0.03600077f
            : (e == 4) ? 0.00759875f
            :            0.00102838f;
    return ((unsigned)d > 10u) ? 0.0f : w;
}

// 1-ulp v_rcp_f32 based sigmoid: feeds a 121-tap weighted average, precision ample.
__device__ __forceinline__ float fast_sigmoid(float x) {
    return __builtin_amdgcn_rcpf(1.0f + __expf(-x));
}

// Branchless zero-padded fetch of (sigmoid(pred), target) at (row, c):
// always load from a clamped (valid) address, multiply by 0/1 in-bounds mask.
// No exec-mask divergence, no branches.
__device__ __forceinline__ void load_pt(const float* __restrict__ P,
                                        const float* __restrict__ T,
                                        int rowbase, bool rowOk, int c,
                                        float& p, float& t) {
    int cc = c < 0 ? 0 : (c > kW - 1 ? kW - 1 : c);      // v_med3
    float m = (rowOk && ((unsigned)c < (unsigned)kW)) ? 1.0f : 0.0f;
    float pr = P[rowbase + cc];
    float tr = T[rowbase + cc];
    p = m * fast_sigmoid(pr);   // sigmoid BEFORE the pad mask (pad is 0, not 0.5)
    t = m * tr;
}

__global__ void __launch_bounds__(kBlockThreads)
ssim_tile_kernel(const float* __restrict__ pred, const float* __restrict__ target,
                 float* __restrict__ partial, float* __restrict__ gout, int use_atomic)
{
    // Per-wave staging of horizontally-blurred mid tiles: 5 channels x 32 rows x 16 cols
    __shared__ float smid[kWavesPerBlock][5][32 * 16];
    __shared__ float sred[kBlockThreads];

    const int tid    = threadIdx.x;
    const int wv     = tid >> 5;   // wave32
    const int lane   = tid & 31;
    const int laneLo = lane & 15;  // M (A) / N (B,C,D)
    const int laneHi = lane >> 4;  // selects K half

    const int tile = blockIdx.x * kWavesPerBlock + wv;
    const int img  = tile >> 10;         // / kTilesPerImg
    const int rem  = tile & 1023;
    const int r0   = (rem >> 5) << 4;
    const int c0   = (rem & 31) << 4;

    const float* __restrict__ P = pred   + (size_t)img * (kH * kW);
    const float* __restrict__ T = target + (size_t)img * (kH * kW);

    // Band coefficients per K-slice: element [K = s*4 + 2*laneHi + v][idx = laneLo],
    // value g[K - idx]. Identical mapping serves as B (Gh) in the horizontal pass
    // and as A (Gv) in the vertical pass.
    v2f gco[7];
    #pragma unroll
    for (int s = 0; s < 7; ++s) {
        int base = s * 4 + 2 * laneHi - laneLo;
        gco[s].x = gtap(base + 0);
        gco[s].y = gtap(base + 1);
    }

    const v8f vzero = {0.f, 0.f, 0.f, 0.f, 0.f, 0.f, 0.f, 0.f};

    // ---------------- Horizontal pass: Mid = In(16x28) x Gh(28x16) ----------------
    #pragma unroll
    for (int b = 0; b < 2; ++b) {
        v8f acc[5];
        #pragma unroll
        for (int ch = 0; ch < 5; ++ch) acc[ch] = vzero;

        const int  row     = r0 - 5 + b * 16 + laneLo;   // A: M = laneLo
        const bool rowOk   = (unsigned)row < (unsigned)kH;
        const int  rowCl   = row < 0 ? 0 : (row > kH - 1 ? kH - 1 : row);
        const int  rowbase = rowCl * kW;

        #pragma unroll
        for (int s = 0; s < 7; ++s) {
            const int cb = c0 - 5 + s * 4 + 2 * laneHi;  // A: K = s*4 + 2*laneHi + v
            float p0, t0, p1, t1;
            load_pt(P, T, rowbase, rowOk, cb,     p0, t0);
            load_pt(P, T, rowbase, rowOk, cb + 1, p1, t1);
            v2f a[5] = { {p0, p1}, {t0, t1}, {p0 * p0, p1 * p1},
                         {t0 * t0, t1 * t1}, {p0 * t0, p1 * t1} };
            #pragma unroll
            for (int ch = 0; ch < 5; ++ch)
                acc[ch] = __builtin_amdgcn_wmma_f32_16x16x4_f32(
                    false, a[ch], false, gco[s], (short)0, acc[ch], false, false);
        }
        // D layout: VGPR v -> mid row (b*16 + v + 8*laneHi), col laneLo
        #pragma unroll
        for (int ch = 0; ch < 5; ++ch)
            #pragma unroll
            for (int v = 0; v < 8; ++v)
                smid[wv][ch][(b * 16 + v + 8 * laneHi) * 16 + laneLo] = acc[ch][v];
    }

    // ---------------- Vertical pass: Out = Gv(16x28) x Mid(28x16) ----------------
    v8f acc[5];
    #pragma unroll
    for (int ch = 0; ch < 5; ++ch) acc[ch] = vzero;

    #pragma unroll
    for (int s = 0; s < 7; ++s) {
        const int mrow = s * 4 + 2 * laneHi;        // B: K = mrow + v, N = laneLo
        #pragma unroll
        for (int ch = 0; ch < 5; ++ch) {
            v2f bm;
            bm.x = smid[wv][ch][(mrow + 0) * 16 + laneLo];
            bm.y = smid[wv][ch][(mrow + 1) * 16 + laneLo];
            acc[ch] = __builtin_amdgcn_wmma_f32_16x16x4_f32(
                false, gco[s], false, bm, (short)0, acc[ch], false, false);
        }
    }

    // ---------------- SSIM map + tile-local sum ----------------
    float lsum = 0.0f;
    #pragma unroll
    for (int v = 0; v < 8; ++v) {
        float mu1 = acc[0][v], mu2 = acc[1][v];
        float bp2 = acc[2][v], bt2 = acc[3][v], bpt = acc[4][v];
        float mu1sq = mu1 * mu1, mu2sq = mu2 * mu2, mu12 = mu1 * mu2;
        float s1 = bp2 - mu1sq, s2 = bt2 - mu2sq, s12 = bpt - mu12;
        float num = (2.0f * mu12 + kC1) * (2.0f * s12 + kC2);
        float den = (mu1sq + mu2sq + kC1) * (s1 + s2 + kC2);
        lsum += 1.0f - num * __builtin_amdgcn_rcpf(den);
    }

    sred[tid] = lsum;
    __syncthreads();
    if (tid == 0) {
        float bsum = 0.0f;
        for (int i = 0; i < kBlockThreads; ++i) bsum += sred[i];  // fixed order
        bsum *= kInvTotal;
        if (use_atomic) atomicAdd(gout, bsum);
        else            partial[blockIdx.x] = bsum;
    }
}

__global__ void __launch_bounds__(256)
ssim_reduce_kernel(const float* __restrict__ partial, float* __restrict__ out, int n)
{
    __shared__ float s[256];
    const int t = threadIdx.x;
    const int chunk = (n + 255) / 256;
    int lo = t * chunk;
    int hi = lo + chunk; if (hi > n) hi = n;
    float a = 0.0f;
    for (int i = lo; i < hi; ++i) a += partial[i];   // fixed per-thread order
    s[t] = a;
    __syncthreads();
    if (t == 0) {
        float r = 0.0f;
        for (int i = 0; i < 256; ++i) r += s[i];     // fixed order
        out[0] = r;
    }
}

__global__ void ssim_zero_out(float* __restrict__ out)
{
    if (threadIdx.x == 0) out[0] = 0.0f;
}

extern "C" void kernel_launch(void* const* d_in, const int* in_sizes, int n_in,
                              void* d_out, int out_size, void* d_ws, size_t ws_size,
                              hipStream_t stream)
{
    (void)in_sizes; (void)n_in; (void)out_size;
    const float* pred   = (const float*)d_in[0];
    const float* target = (const float*)d_in[1];
    float* out = (float*)d_out;

    if (ws_size >= (size_t)kBlocks * sizeof(float)) {
        float* partial = (float*)d_ws;
        ssim_tile_kernel<<<kBlocks, kBlockThreads, 0, stream>>>(pred, target, partial, out, 0);
        ssim_reduce_kernel<<<1, 256, 0, stream>>>(partial, out, kBlocks);
    } else {
        ssim_zero_out<<<1, 32, 0, stream>>>(out);
        ssim_tile_kernel<<<kBlocks, kBlockThreads, 0, stream>>>(pred, target, nullptr, out, 1);
    }
}